// SeqAttention_87986700026302
// MI455X (gfx1250) — compile-verified
//
#include <hip/hip_runtime.h>
#include <hip/hip_bf16.h>

// Problem dims (fixed by the reference)
#define BATCH 8
#define SEQ   256
#define FEAT  6272
#define UNITS 32

typedef float v2f __attribute__((ext_vector_type(2)));
typedef float v8f __attribute__((ext_vector_type(8)));

// ---------------------------------------------------------------------------
// One wave computes one 16x16 f32 tile of C = A(16xK, row-major, lda) *
// B(Kx16, row-major, ldb) using V_WMMA_F32_16X16X4_F32.
//
// ISA operand layouts (CDNA5 ISA §7.12.2):
//   A 16x4 (2 VGPRs): lane l, half h = l>>4, m = l&15:
//       v0 = A[m][k + 2h],  v1 = A[m][k + 2h + 1]
//   B 4x16 (2 VGPRs): lane l, half h, n = l&15:
//       v0 = B[k + 2h][n],  v1 = B[k + 2h + 1][n]
//   C/D 16x16 (8 VGPRs): element i of v8f -> row M = i + 8h, col N = l&15
// ---------------------------------------------------------------------------
__device__ inline v8f wmma_f32_tile(const float* __restrict__ A, int lda,
                                    const float* __restrict__ B, int ldb,
                                    int K) {
  v8f c = {0.f, 0.f, 0.f, 0.f, 0.f, 0.f, 0.f, 0.f};
  const int lane = threadIdx.x & 31;
  const int half = lane >> 4;   // 0 or 1
  const int l15  = lane & 15;

  const float* arow = A + (size_t)l15 * lda + 2 * half;       // A[m][2h + k]
  const float* bcol = B + l15 + (size_t)(2 * half) * ldb;     // B[2h + k][n]

  for (int k = 0; k < K; k += 4) {
    v2f av, bv;
    av.x = arow[0];
    av.y = arow[1];
    bv.x = bcol[0];
    bv.y = bcol[ldb];
    // gfx1250 prefetch path: pull B rows ~32 k-steps ahead into cache
    __builtin_prefetch(bcol + (size_t)128 * ldb, 0, 1);
    // 8-arg form: (neg_a, A, neg_b, B, c_mod, C, reuse_a, reuse_b)
    c = __builtin_amdgcn_wmma_f32_16x16x4_f32(
        /*neg_a=*/false, av, /*neg_b=*/false, bv,
        /*c_mod=*/(short)0, c, /*reuse_a=*/false, /*reuse_b=*/false);
    arow += 4;
    bcol += (size_t)4 * ldb;
  }
  return c;
}

__device__ inline void wmma_store_tile(float* __restrict__ C, int ldc, v8f c) {
  const int lane  = threadIdx.x & 31;
  const int n     = lane & 15;
  const int mbase = (lane >> 4) * 8;
#pragma unroll
  for (int i = 0; i < 8; ++i) {
    C[(size_t)(mbase + i) * ldc + n] = c[i];
  }
}

// ---------------------------------------------------------------------------
// Kernel A: p[b*L + l, u] = sum_f inputs[b*L + l, f] * Wt[f, u]
// Grid: (UNITS/16, (BATCH*SEQ)/16), block: 32 (one wave per 16x16 tile)
// ---------------------------------------------------------------------------
__global__ void seqattn_proj(const float* __restrict__ inputs,
                             const float* __restrict__ Wt,
                             float* __restrict__ p) {
  const int ftile = blockIdx.x;  // 0..1   (N = 32)
  const int mtile = blockIdx.y;  // 0..127 (M = 2048)
  const float* A = inputs + (size_t)mtile * 16 * FEAT;
  const float* B = Wt + ftile * 16;
  v8f c = wmma_f32_tile(A, FEAT, B, UNITS, FEAT);
  wmma_store_tile(p + (size_t)mtile * 16 * UNITS + ftile * 16, UNITS, c);
}

// ---------------------------------------------------------------------------
// Kernel B: for each (b, q): alpha[k] = sigmoid(Wa . tanh(p_q + p_k + bh) + ba)
//           a[b,q,:] = softmax_k(alpha)
// Grid: (SEQ, BATCH), block: SEQ = 256 threads (thread = one key k)
// ---------------------------------------------------------------------------
__global__ void seqattn_weights(const float* __restrict__ p,
                                const float* __restrict__ bh,
                                const float* __restrict__ Wa,
                                const float* __restrict__ ba,
                                float* __restrict__ a) {
  const int q = blockIdx.x;
  const int b = blockIdx.y;
  const int k = threadIdx.x;

  __shared__ float pq[UNITS];   // p[b,q,:] + bh
  __shared__ float wa[UNITS];
  __shared__ float red[SEQ];

  if (threadIdx.x < UNITS) {
    pq[threadIdx.x] = p[((size_t)b * SEQ + q) * UNITS + threadIdx.x] + bh[threadIdx.x];
    wa[threadIdx.x] = Wa[threadIdx.x];
  }
  __syncthreads();

  const float* pk = p + ((size_t)b * SEQ + k) * UNITS;
  float s = ba[0];
#pragma unroll
  for (int u = 0; u < UNITS; ++u) {
    s += wa[u] * tanhf(pq[u] + pk[u]);
  }
  const float alpha = 1.0f / (1.0f + expf(-s));   // sigmoid, in (0,1)

  // softmax over k; alpha bounded in (0,1) so no max-subtraction needed
  const float e = expf(alpha);
  red[k] = e;
  __syncthreads();
  for (int off = SEQ / 2; off > 0; off >>= 1) {
    if (k < off) red[k] += red[k + off];
    __syncthreads();
  }
  const float total = red[0];
  a[(((size_t)b * SEQ) + q) * SEQ + k] = e / total;
}

// ---------------------------------------------------------------------------
// Kernel C: out[b, q, f] = sum_k a[b,q,k] * inputs[b,k,f]
// Per batch: C(256 x 6272) = A(256x256) * X(256x6272)
// Grid: (FEAT/16 = 392, SEQ/16 = 16, BATCH), block: 32 (one wave per tile)
// ---------------------------------------------------------------------------
__global__ void seqattn_context(const float* __restrict__ a,
                                const float* __restrict__ inputs,
                                float* __restrict__ out) {
  const int ftile = blockIdx.x;  // 0..391
  const int qtile = blockIdx.y;  // 0..15
  const int b     = blockIdx.z;  // 0..7

  const float* A = a + (size_t)b * SEQ * SEQ + (size_t)qtile * 16 * SEQ;
  const float* B = inputs + (size_t)b * SEQ * FEAT + ftile * 16;
  float* C = out + (size_t)b * SEQ * FEAT + (size_t)qtile * 16 * FEAT + ftile * 16;

  v8f c = wmma_f32_tile(A, SEQ, B, FEAT, SEQ);
  wmma_store_tile(C, FEAT, c);
}

// ---------------------------------------------------------------------------
extern "C" void kernel_launch(void* const* d_in, const int* in_sizes, int n_in,
                              void* d_out, int out_size, void* d_ws, size_t ws_size,
                              hipStream_t stream) {
  const float* inputs = (const float*)d_in[0];  // [8, 256, 6272]
  const float* Wt     = (const float*)d_in[1];  // [6272, 32]
  const float* bh     = (const float*)d_in[2];  // [32]
  const float* Wa     = (const float*)d_in[3];  // [32, 1]
  const float* ba     = (const float*)d_in[4];  // [1]
  float* out = (float*)d_out;                   // [8, 256, 6272]

  float* p = (float*)d_ws;                      // [2048, 32]  = 256 KB
  float* a = p + (size_t)BATCH * SEQ * UNITS;   // [8,256,256] = 2 MB

  // p = inputs @ Wt
  dim3 gA(UNITS / 16, (BATCH * SEQ) / 16);
  seqattn_proj<<<gA, 32, 0, stream>>>(inputs, Wt, p);

  // a = softmax_k(sigmoid(Wa . tanh(p_q + p_k + bh) + ba))
  dim3 gB(SEQ, BATCH);
  seqattn_weights<<<gB, SEQ, 0, stream>>>(p, bh, Wa, ba, a);

  // out = a @ inputs
  dim3 gC(FEAT / 16, SEQ / 16, BATCH);
  seqattn_context<<<gC, 32, 0, stream>>>(a, inputs, out);
}